// ReLU_class3_norm1_5239860101430
// MI455X (gfx1250) — compile-verified
//
#include <hip/hip_runtime.h>

// CDNA5 / gfx1250 fused MLP+BN(batch-stats)+ReLU pipeline.
// 4 streaming passes (BN batch statistics force global syncs); x is L2-resident
// after pass 1 (128MB < 192MB L2). Matmuls run on V_WMMA_F32_16X16X4_F32,
// inter-layer tile re-layout goes through per-wave LDS slabs.

typedef float v2f __attribute__((ext_vector_type(2)));
typedef float v8f __attribute__((ext_vector_type(8)));

#define BN_EPS 1e-5f

__device__ __forceinline__ v8f wmma4(v2f a, v2f b, v8f c) {
  // D = A(16x4) * B(4x16) + C, fp32, wave32
  return __builtin_amdgcn_wmma_f32_16x16x4_f32(false, a, false, b, (short)0, c,
                                               false, false);
}

__device__ __forceinline__ v2f ldg2(const float* p) {
  return *reinterpret_cast<const v2f*>(p);
}

struct BNC {
  float a_lo, b_lo, a_hi, b_hi;
};

// Per-lane BN fold: p = max(0, a*z + b) for feature n (lo) and n+16 (hi).
__device__ __forceinline__ BNC bn_coefs(const float* __restrict__ stats,
                                        const float* __restrict__ g,
                                        const float* __restrict__ be,
                                        float invN, int n) {
  BNC c;
  {
    float m = stats[n] * invN;
    float v = fmaxf(stats[32 + n] * invN - m * m, 0.f);
    float a = g[n] * rsqrtf(v + BN_EPS);
    c.a_lo = a;
    c.b_lo = be[n] - a * m;
  }
  {
    int f = n + 16;
    float m = stats[f] * invN;
    float v = fmaxf(stats[32 + f] * invN - m * m, 0.f);
    float a = g[f] * rsqrtf(v + BN_EPS);
    c.a_hi = a;
    c.b_hi = be[f] - a * m;
  }
  return c;
}

__global__ void zero_stats_kernel(float* __restrict__ s) {
  int i = threadIdx.x;
  if (i < 192) s[i] = 0.f;
}

// STAGE==1: compute z1, accumulate its per-feature sum/sumsq into stats1.
// STAGE==2: z1->BN1+ReLU->z2, accumulate stats2.
// STAGE==3: ...->z3, accumulate stats3.
// STAGE==4: full forward, write out[N].
template <int STAGE>
__global__ __launch_bounds__(256) void mlp_stage(
    const float* __restrict__ x,
    const float* __restrict__ W1, const float* __restrict__ b1,
    const float* __restrict__ g1, const float* __restrict__ be1,
    const float* __restrict__ W2, const float* __restrict__ b2,
    const float* __restrict__ g2, const float* __restrict__ be2,
    const float* __restrict__ W3, const float* __restrict__ b3,
    const float* __restrict__ g3, const float* __restrict__ be3,
    const float* __restrict__ W4, const float* __restrict__ b4,
    float* __restrict__ stats1, float* __restrict__ stats2,
    float* __restrict__ stats3, float* __restrict__ out, int nTiles,
    float invN) {
  // Per-wave 16x32 staging tile, rows padded to 34 floats (136B) to spread
  // LDS banks for the strided b64 A-fragment reads.
  __shared__ float lds[8 * 16 * 34];
  const int lane = threadIdx.x & 31;
  const int n = lane & 15;   // column / row-in-tile selector
  const int hi = lane >> 4;  // which half of the wave
  float* slab = lds + (threadIdx.x >> 5) * (16 * 34);

  // ---- loop-invariant weight fragments (B-matrix layout) ----
  // B[k][j] = W[j][k]; lane n holds cols 4c+2*hi..+1 of feature row 16*t+n.
  v2f bw1[2][4];
#pragma unroll
  for (int t = 0; t < 2; ++t)
#pragma unroll
    for (int c = 0; c < 4; ++c)
      bw1[t][c] = ldg2(W1 + (16 * t + n) * 16 + 4 * c + 2 * hi);
  const float bias1lo = b1[n], bias1hi = b1[16 + n];

  v2f bw2[2][8], bw3[2][8];
  float bias2lo = 0.f, bias2hi = 0.f, bias3lo = 0.f, bias3hi = 0.f;
  float w4lo = 0.f, w4hi = 0.f, bias4 = 0.f;
  BNC c1 = {}, c2 = {}, c3 = {};

  if (STAGE >= 2) {
    c1 = bn_coefs(stats1, g1, be1, invN, n);
#pragma unroll
    for (int t = 0; t < 2; ++t)
#pragma unroll
      for (int c = 0; c < 8; ++c)
        bw2[t][c] = ldg2(W2 + (16 * t + n) * 32 + 4 * c + 2 * hi);
    bias2lo = b2[n];
    bias2hi = b2[16 + n];
  }
  if (STAGE >= 3) {
    c2 = bn_coefs(stats2, g2, be2, invN, n);
#pragma unroll
    for (int t = 0; t < 2; ++t)
#pragma unroll
      for (int c = 0; c < 8; ++c)
        bw3[t][c] = ldg2(W3 + (16 * t + n) * 32 + 4 * c + 2 * hi);
    bias3lo = b3[n];
    bias3hi = b3[16 + n];
  }
  if (STAGE >= 4) {
    c3 = bn_coefs(stats3, g3, be3, invN, n);
    w4lo = W4[n];
    w4hi = W4[16 + n];
    bias4 = b4[0];
  }

  float slo = 0.f, qlo = 0.f, shi = 0.f, qhi = 0.f;

  const int waveId = (int)((blockIdx.x * blockDim.x + threadIdx.x) >> 5);
  const int nW = (int)((gridDim.x * blockDim.x) >> 5);

  for (int tile = waveId; tile < nTiles; tile += nW) {
    // ---- layer 1: A-frag straight from global x (row = n, cols 4c+2hi) ----
    const float* xr = x + (size_t)(tile * 16 + n) * 16;
    v8f d0, d1;
#pragma unroll
    for (int e = 0; e < 8; ++e) {
      d0[e] = bias1lo;
      d1[e] = bias1hi;
    }
#pragma unroll
    for (int c = 0; c < 4; ++c) {
      v2f a = ldg2(xr + 4 * c + 2 * hi);
      d0 = wmma4(a, bw1[0][c], d0);
      d1 = wmma4(a, bw1[1][c], d1);
    }
    if (STAGE == 1) {
#pragma unroll
      for (int e = 0; e < 8; ++e) {
        slo += d0[e];
        qlo += d0[e] * d0[e];
        shi += d1[e];
        qhi += d1[e] * d1[e];
      }
      continue;
    }

    // ---- BN1 + ReLU, stage tile to LDS (D layout -> row-major) ----
#pragma unroll
    for (int e = 0; e < 8; ++e) {
      d0[e] = fmaxf(fmaf(d0[e], c1.a_lo, c1.b_lo), 0.f);
      d1[e] = fmaxf(fmaf(d1[e], c1.a_hi, c1.b_hi), 0.f);
    }
#pragma unroll
    for (int r = 0; r < 8; ++r) {
      slab[(r + 8 * hi) * 34 + n] = d0[r];
      slab[(r + 8 * hi) * 34 + 16 + n] = d1[r];
    }

    // ---- layer 2 ----
    v8f e0, e1;
#pragma unroll
    for (int e = 0; e < 8; ++e) {
      e0[e] = bias2lo;
      e1[e] = bias2hi;
    }
#pragma unroll
    for (int c = 0; c < 8; ++c) {
      v2f a = *reinterpret_cast<const v2f*>(&slab[n * 34 + 4 * c + 2 * hi]);
      e0 = wmma4(a, bw2[0][c], e0);
      e1 = wmma4(a, bw2[1][c], e1);
    }
    if (STAGE == 2) {
#pragma unroll
      for (int e = 0; e < 8; ++e) {
        slo += e0[e];
        qlo += e0[e] * e0[e];
        shi += e1[e];
        qhi += e1[e] * e1[e];
      }
      continue;
    }

    // ---- BN2 + ReLU, restage ----
#pragma unroll
    for (int e = 0; e < 8; ++e) {
      e0[e] = fmaxf(fmaf(e0[e], c2.a_lo, c2.b_lo), 0.f);
      e1[e] = fmaxf(fmaf(e1[e], c2.a_hi, c2.b_hi), 0.f);
    }
#pragma unroll
    for (int r = 0; r < 8; ++r) {
      slab[(r + 8 * hi) * 34 + n] = e0[r];
      slab[(r + 8 * hi) * 34 + 16 + n] = e1[r];
    }

    // ---- layer 3 ----
    v8f f0, f1;
#pragma unroll
    for (int e = 0; e < 8; ++e) {
      f0[e] = bias3lo;
      f1[e] = bias3hi;
    }
#pragma unroll
    for (int c = 0; c < 8; ++c) {
      v2f a = *reinterpret_cast<const v2f*>(&slab[n * 34 + 4 * c + 2 * hi]);
      f0 = wmma4(a, bw3[0][c], f0);
      f1 = wmma4(a, bw3[1][c], f1);
    }
    if (STAGE == 3) {
#pragma unroll
      for (int e = 0; e < 8; ++e) {
        slo += f0[e];
        qlo += f0[e] * f0[e];
        shi += f1[e];
        qhi += f1[e] * f1[e];
      }
      continue;
    }

    // ---- BN3 + ReLU + layer 4 (32 -> 1) cross-lane dot ----
#pragma unroll
    for (int e = 0; e < 8; ++e) {
      f0[e] = fmaxf(fmaf(f0[e], c3.a_lo, c3.b_lo), 0.f);
      f1[e] = fmaxf(fmaf(f1[e], c3.a_hi, c3.b_hi), 0.f);
    }
#pragma unroll
    for (int r = 0; r < 8; ++r) {
      // lane holds row (r + 8*hi), features n and 16+n
      float t = f0[r] * w4lo + f1[r] * w4hi;
      t += __shfl_xor(t, 1, 32);
      t += __shfl_xor(t, 2, 32);
      t += __shfl_xor(t, 4, 32);
      t += __shfl_xor(t, 8, 32);
      if (n == 0) out[tile * 16 + r + 8 * hi] = t + bias4;
    }
  }

  // ---- flush per-wave stats ----
  if (STAGE <= 3) {
    slo += __shfl_xor(slo, 16, 32);
    qlo += __shfl_xor(qlo, 16, 32);
    shi += __shfl_xor(shi, 16, 32);
    qhi += __shfl_xor(qhi, 16, 32);
    float* st = (STAGE == 1) ? stats1 : (STAGE == 2) ? stats2 : stats3;
    if (lane < 16) {
      atomicAdd(st + n, slo);        // sum, features 0..15
      atomicAdd(st + 16 + n, shi);   // sum, features 16..31
      atomicAdd(st + 32 + n, qlo);   // sumsq, features 0..15
      atomicAdd(st + 48 + n, qhi);   // sumsq, features 16..31
    }
  }
}

extern "C" void kernel_launch(void* const* d_in, const int* in_sizes, int n_in,
                              void* d_out, int out_size, void* d_ws,
                              size_t ws_size, hipStream_t stream) {
  (void)n_in;
  (void)out_size;
  (void)ws_size;
  const float* x = (const float*)d_in[0];
  const float* W1 = (const float*)d_in[1];
  const float* b1 = (const float*)d_in[2];
  const float* g1 = (const float*)d_in[3];
  const float* be1 = (const float*)d_in[4];
  const float* W2 = (const float*)d_in[5];
  const float* b2 = (const float*)d_in[6];
  const float* g2 = (const float*)d_in[7];
  const float* be2 = (const float*)d_in[8];
  const float* W3 = (const float*)d_in[9];
  const float* b3 = (const float*)d_in[10];
  const float* g3 = (const float*)d_in[11];
  const float* be3 = (const float*)d_in[12];
  const float* W4 = (const float*)d_in[13];
  const float* b4 = (const float*)d_in[14];

  const int N = in_sizes[0] / 16;
  const int nTiles = N / 16;  // N = 2^21, exact
  const float invN = 1.0f / (float)N;

  float* stats = (float*)d_ws;  // [s1:64][s2:64][s3:64]
  float* out = (float*)d_out;

  zero_stats_kernel<<<1, 256, 0, stream>>>(stats);

  dim3 blk(256);   // 8 waves (wave32) per workgroup
  dim3 grd(512);   // persistent-ish grid; each wave grid-strides tiles

  mlp_stage<1><<<grd, blk, 0, stream>>>(x, W1, b1, g1, be1, W2, b2, g2, be2,
                                        W3, b3, g3, be3, W4, b4, stats,
                                        stats + 64, stats + 128, out, nTiles,
                                        invN);
  mlp_stage<2><<<grd, blk, 0, stream>>>(x, W1, b1, g1, be1, W2, b2, g2, be2,
                                        W3, b3, g3, be3, W4, b4, stats,
                                        stats + 64, stats + 128, out, nTiles,
                                        invN);
  mlp_stage<3><<<grd, blk, 0, stream>>>(x, W1, b1, g1, be1, W2, b2, g2, be2,
                                        W3, b3, g3, be3, W4, b4, stats,
                                        stats + 64, stats + 128, out, nTiles,
                                        invN);
  mlp_stage<4><<<grd, blk, 0, stream>>>(x, W1, b1, g1, be1, W2, b2, g2, be2,
                                        W3, b3, g3, be3, W4, b4, stats,
                                        stats + 64, stats + 128, out, nTiles,
                                        invN);
}